// SqRL_47682726920454
// MI455X (gfx1250) — compile-verified
//
#include <hip/hip_runtime.h>
#include <hip/hip_bf16.h>
#include <stdint.h>

// Ring-perimeter gather: x (32,3,512,512) fp32 -> out (32,3,256,2048) fp32.
// Memory-bound, 0 FLOPs: ~100 MB read + ~201 MB written => ~13 us @ 23.3 TB/s.
// Split by source orientation so EVERY global access is full-cacheline
// coalesced (no 4B-per-128B-line column gathers on the L2->WGP path):
//   Kernel R: row-sourced parts (corners + top/bottom edges) from async-staged
//             rows i and j (ASYNCcnt path).
//   Kernel C: column-sourced parts (left/right edges). The 512x32 padded-LDS
//             tile stage is a single Tensor Data Mover descriptor per block
//             (TENSORcnt path, pad_enable builds the bank-conflict-free
//             stride-33 layout in hardware); async-b32 loop as fallback.
// NT stores keep the 201 MB output stream from evicting the L2-resident input.

typedef float v4f  __attribute__((ext_vector_type(4)));
typedef unsigned int u32x4 __attribute__((ext_vector_type(4)));
typedef int   i32x4 __attribute__((ext_vector_type(4)));
typedef int   i32x8 __attribute__((ext_vector_type(8)));

#define RG_H   512
#define TILE_C 32
#define PAD    33        // gcd(33,64)=1 -> conflict-free LDS column reads

__device__ __forceinline__ void async_g2l_b64(uint32_t lds_off, const void* g) {
  asm volatile("global_load_async_to_lds_b64 %0, %1, off"
               :: "v"(lds_off), "v"((uint64_t)(uintptr_t)g) : "memory");
}
__device__ __forceinline__ void async_g2l_b32(uint32_t lds_off, const void* g) {
  asm volatile("global_load_async_to_lds_b32 %0, %1, off"
               :: "v"(lds_off), "v"((uint64_t)(uintptr_t)g) : "memory");
}
__device__ __forceinline__ void wait_async0() {
  asm volatile("s_wait_asynccnt 0x0" ::: "memory");
}

// ---------------------------------------------------------------------------
// Kernel R: one block per (plane, ring). Emits all output columns k except the
// two column-sourced segments [766-dif,767+dif) and [1788-dif,1789+dif).
// ---------------------------------------------------------------------------
__global__ __launch_bounds__(256)
void SqRL_rows(const float* __restrict__ x, float* __restrict__ out) {
  __shared__ float smem[2 * RG_H];                 // rowI | rowJ   (4 KB)
  const int t   = threadIdx.x;
  const int dif = blockIdx.x;                      // 0..255
  const int p   = blockIdx.y;                      // 0..95
  const int i   = 255 - dif;
  const int j   = 256 + dif;
  const float* plane = x + (size_t)p * (RG_H * RG_H);
  const uint32_t lds0 = (uint32_t)(uintptr_t)(void*)smem;

  // async stage rows i and j (contiguous, 2 floats/lane)
  async_g2l_b64(lds0 + 8u * t,          plane + (size_t)i * RG_H + 2 * t);
  async_g2l_b64(lds0 + 2048u + 8u * t,  plane + (size_t)j * RG_H + 2 * t);
  wait_async0();
  __syncthreads();

  const int c0l = 766 - dif,  c0r = 767 + dif;     // right edge (kernel C's)
  const int c1l = 1788 - dif, c1r = 1789 + dif;    // left edge  (kernel C's)
  const size_t ob = ((size_t)(p * 256 + dif)) * 2048;

#pragma unroll
  for (int h = 0; h < 2; ++h) {
    const int k0 = h * 1024 + 4 * t;
    const bool in0 = (k0 >= c0l) & (k0 + 4 <= c0r);
    const bool in1 = (k0 >= c1l) & (k0 + 4 <= c1r);
    if (in0 | in1) continue;                       // fully column-sourced

    float v[4];
#pragma unroll
    for (int e = 0; e < 4; ++e) {
      int k = k0 + e;
      if (k >= 2044) k -= 2044;                    // 4-entry wraparound copy
      int a = i;                                   // corner A = x[i][i]
      if (k >= i)          a = k;                  // top edge : rowI[k]
      if (k >= j)          a = j;                  // corner B = x[i][j]
      if (k >= 767 + dif)  a = 512 + j;            // corner C = x[j][j]
      if (k >= 1277 - dif) a = 2045 - k;           // bottom edge (reversed)
      if (k >= 1278 + dif) a = 512 + i;            // corner D = x[j][i]
      if (k >= 1789 + dif) a = i;                  // trailing corner-A run
      v[e] = smem[a];
    }

    const bool ov0 = (k0 < c0r) & (k0 + 4 > c0l);
    const bool ov1 = (k0 < c1r) & (k0 + 4 > c1l);
    if (!(ov0 | ov1)) {
      v4f q = {v[0], v[1], v[2], v[3]};
      __builtin_nontemporal_store(q, (v4f*)(out + ob + k0));
    } else {                                       // segment boundary: mask
#pragma unroll
      for (int e = 0; e < 4; ++e) {
        const int k = k0 + e;
        const bool col = ((k >= c0l) & (k < c0r)) | ((k >= c1l) & (k < c1r));
        if (!col) __builtin_nontemporal_store(v[e], out + ob + k);
      }
    }
  }
}

// ---------------------------------------------------------------------------
// Kernel C: one block per (plane, 32-column tile). Stages x[0:512, c0:c0+32]
// into pad-33 LDS, then emits the column-sourced edge segments:
//   c >= 256: column j of ring dif=c-256 : out[766-dif+m]  = x[i+m][j]
//   c <  256: column i of ring dif=255-c : out[1788-dif+m] = x[j-m][i]
// ---------------------------------------------------------------------------
__global__ __launch_bounds__(256)
void SqRL_cols(const float* __restrict__ x, float* __restrict__ out) {
  __shared__ float smem[RG_H * PAD];               // 512 x 32 tile, 66 KB
  const int t  = threadIdx.x;
  const int c0 = blockIdx.x * TILE_C;
  const int p  = blockIdx.y;
  const float* plane = x + (size_t)p * (RG_H * RG_H);
  const uint32_t lds0 = (uint32_t)(uintptr_t)(void*)smem;
  const int l = t & 31;                            // lane
  const int w = t >> 5;                            // wave in block

  // ---- phase 1: stage the tile --------------------------------------------
#if __has_builtin(__builtin_amdgcn_tensor_load_to_lds)
  // One TDM descriptor moves the whole 64 KB tile: 2D tensor, 4B elements,
  // tile 32 x 512, row stride 512 elements, LDS pad of 1 dword after every
  // 32 dwords (pad_interval enc 4 = 32 dw, pad_amount enc 0 = 1 dw) -> the
  // stride-33 layout is built by the DMA engine. D# packing per ISA ch.8.
  if (w == 0) {
    const uint64_t ga = (uint64_t)(uintptr_t)(plane + c0);
    u32x4 g0;
    g0.x = 1u;                                       // count=1, user mode
    g0.y = lds0;                                     // lds_addr (bytes)
    g0.z = (uint32_t)ga;                             // global_addr[31:0]
    g0.w = (uint32_t)((ga >> 32) & 0x01FFFFFFu)      // global_addr[56:32]
         | (2u << 30);                               // type = 2 ("image")
    i32x8 g1;
    g1[0] = (2 << 16)                                // data_size = 4 B
          | (1 << 20)                                // pad_enable
          | (4 << 22);                               // pad_interval = 32 dw
                                                     // pad_amount enc 0 = 1 dw
    g1[1] = (int)(512u << 16);                       // tensor_dim0 = 512
    g1[2] = (int)(512u << 16);                       // tensor_dim1 = 512
    g1[3] = (int)(32u  << 16);                       // tile_dim0   = 32
    g1[4] = 512;                                     // tile_dim1   = 512
    g1[5] = 512;                                     // tensor_dim0_stride
    g1[6] = 0;
    g1[7] = 0;
    i32x4 gz = {0, 0, 0, 0};                         // groups 2/3: unused (2D)
#if defined(__clang_major__) && __clang_major__ >= 23
    i32x8 gz8 = {0, 0, 0, 0, 0, 0, 0, 0};
    __builtin_amdgcn_tensor_load_to_lds(g0, g1, gz, gz, gz8, 0);
#else
    __builtin_amdgcn_tensor_load_to_lds(g0, g1, gz, gz, 0);
#endif
    __builtin_amdgcn_s_wait_tensorcnt((short)0);
  }
#else
  // Fallback: per-lane async copies; each wave op = one contiguous 128B chunk.
  {
    const float* g = plane + (size_t)w * RG_H + c0 + l;   // row r=w, col c0+l
    uint32_t lo = lds0 + (uint32_t)((w * PAD + l) << 2);
#pragma unroll 4
    for (int it = 0; it < 64; ++it) {              // rows r = w + 8*it
      async_g2l_b32(lo, g);
      g  += 8 * RG_H;
      lo += (8 * PAD) << 2;
    }
  }
  wait_async0();
#endif
  __syncthreads();

  // ---- phase 2: column reads from LDS (pad-33 => conflict-free), NT stores
  for (int it = 0; it < 64; ++it) {
    const int cp  = 4 * w + (it & 3);              // wave-uniform tile column
    const int mb  = (it >> 2) * 32;                // wave-uniform m base
    const int c   = c0 + cp;
    const int dif = (c >= 256) ? (c - 256) : (255 - c);
    const int el  = 2 * dif + 1;
    if (mb >= el) continue;                        // wave-uniform early skip
    const int m  = mb + l;
    const int mc = (m < el) ? m : (el - 1);        // clamp for safe LDS read
    const int i  = 255 - dif;
    const int j  = 256 + dif;
    const size_t ob = ((size_t)(p * 256 + dif)) * 2048;
    int k; float v;
    if (c >= 256) { k = 766 - dif + m;  v = smem[(i + mc) * PAD + cp]; }
    else          { k = 1788 - dif + m; v = smem[(j - mc) * PAD + cp]; }
    if (m < el) __builtin_nontemporal_store(v, out + ob + k);
  }
}

extern "C" void kernel_launch(void* const* d_in, const int* in_sizes, int n_in,
                              void* d_out, int out_size, void* d_ws, size_t ws_size,
                              hipStream_t stream) {
  const float* x = (const float*)d_in[0];
  float* out = (float*)d_out;
  const int planes = in_sizes[0] / (RG_H * RG_H);  // 32*3 = 96

  // R first (warms L2 with the planes), C second (hits L2, fills the holes).
  SqRL_rows<<<dim3(RG_H / 2, planes),      256, 0, stream>>>(x, out);
  SqRL_cols<<<dim3(RG_H / TILE_C, planes), 256, 0, stream>>>(x, out);
}